// QuantizerTl_23983097381623
// MI455X (gfx1250) — compile-verified
//
#include <hip/hip_runtime.h>
#include <hip/hip_fp16.h>

#define HIDDEN 4096
#define POD 128
#define ROWLEN (HIDDEN + POD)   /* 4224 fp32 per row */
#define QMAX 7.0f
#define BLOCK 256               /* 8 wave32 waves */
#define NWAVES (BLOCK / 32)

// Outputs concatenated flat (all exposed as fp32 by the harness):
//   [0, rows*HIDDEN)                       : q  (quantized values, as float)
//   [rows*HIDDEN, rows*HIDDEN+rows)        : scales (fp16 value widened to float)
//   [rows*HIDDEN+rows, ... +rows*POD)      : outlier passthrough

__global__ __launch_bounds__(BLOCK) void quantize_rows(const float* __restrict__ x,
                                                       float* __restrict__ out,
                                                       int rows) {
    __shared__ float smem[HIDDEN];        // 16 KB row staging
    __shared__ float wred[NWAVES];        // per-wave partial maxes

    const int row = blockIdx.x;
    const int tid = threadIdx.x;

    const float* rowp  = x + (size_t)row * ROWLEN;
    const float* mainp = rowp + POD;

    // ---- async DMA: stage main[4096] into LDS, 4 x b128 per thread ----
    // Coalesced: chunk c, thread t touches byte (c*4096 + t*16) on both sides
    // (the instruction immediate offset is added to BOTH global and LDS addr).
    unsigned           ldsAddr = (unsigned)(uintptr_t)(&smem[0]) + (unsigned)tid * 16u;
    unsigned long long gAddr   = (unsigned long long)(uintptr_t)mainp + (unsigned long long)tid * 16u;
    asm volatile(
        "global_load_async_to_lds_b128 %0, %1, off\n\t"
        "global_load_async_to_lds_b128 %0, %1, off offset:4096\n\t"
        "global_load_async_to_lds_b128 %0, %1, off offset:8192\n\t"
        "global_load_async_to_lds_b128 %0, %1, off offset:12288"
        :: "v"(ldsAddr), "v"(gAddr) : "memory");

    // ---- outlier passthrough while the DMA is in flight ----
    float* outlierOut = out + (size_t)rows * HIDDEN + (size_t)rows; // fp32-aligned
    if (tid < POD / 4) {
        const float4 v = ((const float4*)rowp)[tid];
        ((float4*)outlierOut)[(size_t)row * (POD / 4) + tid] = v;
    }

    // wait for this wave's async copies, then make LDS visible block-wide
    asm volatile("s_wait_asynccnt 0" ::: "memory");
    __syncthreads();

    // ---- each thread owns 16 contiguous floats of the row ----
    const float4* sp = (const float4*)smem;
    const float4 a0 = sp[tid * 4 + 0];
    const float4 a1 = sp[tid * 4 + 1];
    const float4 a2 = sp[tid * 4 + 2];
    const float4 a3 = sp[tid * 4 + 3];

    float m = fabsf(a0.x);
    m = fmaxf(m, fabsf(a0.y)); m = fmaxf(m, fabsf(a0.z)); m = fmaxf(m, fabsf(a0.w));
    m = fmaxf(m, fabsf(a1.x)); m = fmaxf(m, fabsf(a1.y)); m = fmaxf(m, fabsf(a1.z)); m = fmaxf(m, fabsf(a1.w));
    m = fmaxf(m, fabsf(a2.x)); m = fmaxf(m, fabsf(a2.y)); m = fmaxf(m, fabsf(a2.z)); m = fmaxf(m, fabsf(a2.w));
    m = fmaxf(m, fabsf(a3.x)); m = fmaxf(m, fabsf(a3.y)); m = fmaxf(m, fabsf(a3.z)); m = fmaxf(m, fabsf(a3.w));

    // wave32 butterfly reduction
    #pragma unroll
    for (int off = 16; off > 0; off >>= 1)
        m = fmaxf(m, __shfl_xor(m, off, 32));

    if ((tid & 31) == 0) wred[tid >> 5] = m;
    __syncthreads();

    float mx = wred[0];
    #pragma unroll
    for (int i = 1; i < NWAVES; ++i) mx = fmaxf(mx, wred[i]);

    // scale: fp32 max/7 -> fp16 (CLIP=1.0 multiply is identity) -> back to fp32
    const __half hs = __float2half(mx / QMAX);
    const float  s  = __half2float(hs);
    const float  inv = 1.0f / s;

    // quantize: round-to-nearest-even, clamp [-8, 7], emit as fp32
    #define Q8(v) fminf(7.0f, fmaxf(-8.0f, rintf((v) * inv)))
    float4 q0 = make_float4(Q8(a0.x), Q8(a0.y), Q8(a0.z), Q8(a0.w));
    float4 q1 = make_float4(Q8(a1.x), Q8(a1.y), Q8(a1.z), Q8(a1.w));
    float4 q2 = make_float4(Q8(a2.x), Q8(a2.y), Q8(a2.z), Q8(a2.w));
    float4 q3 = make_float4(Q8(a3.x), Q8(a3.y), Q8(a3.z), Q8(a3.w));
    #undef Q8

    float4* qout = (float4*)out;
    const size_t qbase = (size_t)row * (HIDDEN / 4) + (size_t)tid * 4;
    qout[qbase + 0] = q0;
    qout[qbase + 1] = q1;
    qout[qbase + 2] = q2;
    qout[qbase + 3] = q3;

    if (tid == 0) {
        out[(size_t)rows * HIDDEN + row] = s;   // fp16-rounded scale, widened
    }
}

extern "C" void kernel_launch(void* const* d_in, const int* in_sizes, int n_in,
                              void* d_out, int out_size, void* d_ws, size_t ws_size,
                              hipStream_t stream) {
    (void)n_in; (void)out_size; (void)d_ws; (void)ws_size;
    const float* x = (const float*)d_in[0];
    const int rows = in_sizes[0] / ROWLEN;   // 4 * 2048 = 8192
    quantize_rows<<<rows, BLOCK, 0, stream>>>(x, (float*)d_out, rows);
}